// ResidualVectorQuantizer_3006477107478
// MI455X (gfx1250) — compile-verified
//
#include <hip/hip_runtime.h>
#include <hip/hip_bf16.h>

#define DIM   512
#define KCODES 2048
#define NUM_Q 8
#define BT    (8 * 2048)          // flattened B*T rows
#define RPB   16                  // rows per block
#define STR   516                 // padded LDS row stride (floats) -> bank-conflict-free
#define NTHR  128                 // 4 waves
#define NBLK  (BT / RPB)          // 1024 blocks

typedef __attribute__((ext_vector_type(2))) float v2f;
typedef __attribute__((ext_vector_type(8))) float v8f;

// ---------------- codebook norms: norms[q*K + k] = sum(cb[q][k][:]^2) --------
__global__ __launch_bounds__(128) void rvq_cb_norms(const float* __restrict__ cb,
                                                    float* __restrict__ norms) {
    const int lane = threadIdx.x & 31;
    const int wave = threadIdx.x >> 5;
    const int row  = blockIdx.x * 4 + wave;          // 0 .. NUM_Q*KCODES-1
    const float4* p = (const float4*)(cb + (size_t)row * DIM);
    float s = 0.f;
    #pragma unroll
    for (int i = lane; i < DIM / 4; i += 32) {
        float4 v = p[i];
        s += v.x * v.x + v.y * v.y + v.z * v.z + v.w * v.w;
    }
    #pragma unroll
    for (int m = 16; m >= 1; m >>= 1) s += __shfl_xor(s, m, 32);
    if (lane == 0) norms[row] = s;
}

// ---------------- fused RVQ: all 8 levels, residual tile resident in LDS -----
__global__ __launch_bounds__(NTHR) void rvq_main(const float* __restrict__ x,
                                                 const float* __restrict__ codebooks,
                                                 const float* __restrict__ norms,
                                                 float* __restrict__ outQ,
                                                 float* __restrict__ outIdx,
                                                 float* __restrict__ partial) {
    __shared__ float resid[RPB * STR];               // 33 KB residual tile
    __shared__ float cand_s[4][RPB];
    __shared__ int   cand_i[4][RPB];
    __shared__ int   winIdx[RPB];
    __shared__ float red[NTHR];

    const int tid   = threadIdx.x;
    const int lane  = tid & 31;
    const int wave  = tid >> 5;
    const int nlo   = lane & 15;                     // code column within 16-wide tile
    const int khalf = lane >> 4;                     // K/row-group half selector
    const int m0    = blockIdx.x * RPB;

    // load x tile into LDS (16B-aligned: 516*4 = 2064 = 129*16)
    for (int r = 0; r < RPB; ++r) {
        float4 v = *(const float4*)(x + (size_t)(m0 + r) * DIM + tid * 4);
        *(float4*)(&resid[r * STR + tid * 4]) = v;
    }
    __syncthreads();

    float ssum = 0.f;                                // sum of squared residuals (all levels)
    const float* abase = &resid[nlo * STR + 2 * khalf];

    for (int q = 0; q < NUM_Q; ++q) {
        const float* cbL  = codebooks + (size_t)q * KCODES * DIM;
        const float* nrmL = norms + q * KCODES;

        float best[8]; int bidx[8];
        #pragma unroll
        for (int j = 0; j < 8; ++j) { best[j] = 3.4e38f; bidx[j] = 0; }

        // wave scans codes n0 = wave*16, wave*16+64, ... (16-wide tiles)
        for (int n0 = wave * 16; n0 < KCODES; n0 += 64) {
            const float* bbase = cbL + (size_t)(n0 + nlo) * DIM + 2 * khalf;
            v8f acc0 = {}, acc1 = {}, acc2 = {}, acc3 = {};
            #pragma unroll 4
            for (int d0 = 0; d0 < DIM; d0 += 16) {
                v2f a0 = *(const v2f*)(abase + d0);
                v2f b0 = *(const v2f*)(bbase + d0);
                acc0 = __builtin_amdgcn_wmma_f32_16x16x4_f32(false, a0, false, b0,
                                                             (short)0, acc0, false, false);
                v2f a1 = *(const v2f*)(abase + d0 + 4);
                v2f b1 = *(const v2f*)(bbase + d0 + 4);
                acc1 = __builtin_amdgcn_wmma_f32_16x16x4_f32(false, a1, false, b1,
                                                             (short)0, acc1, false, false);
                v2f a2 = *(const v2f*)(abase + d0 + 8);
                v2f b2 = *(const v2f*)(bbase + d0 + 8);
                acc2 = __builtin_amdgcn_wmma_f32_16x16x4_f32(false, a2, false, b2,
                                                             (short)0, acc2, false, false);
                v2f a3 = *(const v2f*)(abase + d0 + 12);
                v2f b3 = *(const v2f*)(bbase + d0 + 12);
                acc3 = __builtin_amdgcn_wmma_f32_16x16x4_f32(false, a3, false, b3,
                                                             (short)0, acc3, false, false);
            }
            acc0 += acc1; acc2 += acc3; acc0 += acc2;

            const float cn  = nrmL[n0 + nlo];
            const int   myn = n0 + nlo;
            #pragma unroll
            for (int j = 0; j < 8; ++j) {            // row = j + 8*khalf, code = myn
                float s = cn - 2.f * acc0[j];
                if (s < best[j]) { best[j] = s; bidx[j] = myn; }
            }
        }

        // argmin reduce across the 16 code-columns (independent lane halves)
        #pragma unroll
        for (int j = 0; j < 8; ++j) {
            #pragma unroll
            for (int m = 8; m >= 1; m >>= 1) {
                float os = __shfl_xor(best[j], m, 32);
                int   oi = __shfl_xor(bidx[j], m, 32);
                if (os < best[j] || (os == best[j] && oi < bidx[j])) { best[j] = os; bidx[j] = oi; }
            }
        }
        if (nlo == 0) {
            #pragma unroll
            for (int j = 0; j < 8; ++j) {
                cand_s[wave][khalf * 8 + j] = best[j];
                cand_i[wave][khalf * 8 + j] = bidx[j];
            }
        }
        __syncthreads();

        // cross-wave argmin + emit index
        if (tid < RPB) {
            float bs = cand_s[0][tid]; int bi = cand_i[0][tid];
            #pragma unroll
            for (int w = 1; w < 4; ++w) {
                float s2 = cand_s[w][tid]; int i2 = cand_i[w][tid];
                if (s2 < bs || (s2 == bs && i2 < bi)) { bs = s2; bi = i2; }
            }
            winIdx[tid] = bi;
            outIdx[(size_t)(m0 + tid) * NUM_Q + q] = (float)bi;
        }
        __syncthreads();

        // residual -= winner code; accumulate sum((new residual)^2) == level loss numerator
        for (int r = 0; r < RPB; ++r) {
            const float4 c4 = *(const float4*)(cbL + (size_t)winIdx[r] * DIM + tid * 4);
            float4* rp = (float4*)(&resid[r * STR + tid * 4]);
            float4 rv = *rp;
            rv.x -= c4.x; rv.y -= c4.y; rv.z -= c4.z; rv.w -= c4.w;
            *rp = rv;
            ssum += rv.x * rv.x + rv.y * rv.y + rv.z * rv.z + rv.w * rv.w;
        }
        __syncthreads();
    }

    // quantized = x - final residual
    for (int r = 0; r < RPB; ++r) {
        size_t off = (size_t)(m0 + r) * DIM + tid * 4;
        float4 xv = *(const float4*)(x + off);
        float4 rv = *(const float4*)(&resid[r * STR + tid * 4]);
        float4 o; o.x = xv.x - rv.x; o.y = xv.y - rv.y; o.z = xv.z - rv.z; o.w = xv.w - rv.w;
        *(float4*)(outQ + off) = o;
    }

    // deterministic per-block loss partial
    red[tid] = ssum;
    __syncthreads();
    #pragma unroll
    for (int s = 64; s >= 1; s >>= 1) {
        if (tid < s) red[tid] += red[tid + s];
        __syncthreads();
    }
    if (tid == 0) partial[blockIdx.x] = red[0];
}

// ---------------- deterministic final loss reduction -------------------------
__global__ __launch_bounds__(256) void rvq_loss(const float* __restrict__ partial,
                                                float* __restrict__ outLoss) {
    __shared__ float red[256];
    const int tid = threadIdx.x;
    float s = 0.f;
    for (int i = tid; i < NBLK; i += 256) s += partial[i];
    red[tid] = s;
    __syncthreads();
    #pragma unroll
    for (int k = 128; k >= 1; k >>= 1) {
        if (tid < k) red[tid] += red[tid + k];
        __syncthreads();
    }
    // loss = (1/NUM_Q) * sum_levels 1.25 * mean(residual^2)
    if (tid == 0) outLoss[0] = red[0] * (1.25f / ((float)NUM_Q * (float)BT * (float)DIM));
}

extern "C" void kernel_launch(void* const* d_in, const int* in_sizes, int n_in,
                              void* d_out, int out_size, void* d_ws, size_t ws_size,
                              hipStream_t stream) {
    const float* x  = (const float*)d_in[0];     // (8,2048,512) f32
    const float* cb = (const float*)d_in[1];     // (8,2048,512) f32

    float* outQ    = (float*)d_out;                          // 8,388,608 floats
    float* outIdx  = outQ + (size_t)BT * DIM;                // 131,072 floats
    float* outLoss = outIdx + (size_t)BT * NUM_Q;            // 1 float

    float* norms   = (float*)d_ws;                           // 16,384 floats
    float* partial = norms + NUM_Q * KCODES;                 // 1,024 floats

    rvq_cb_norms<<<NUM_Q * KCODES / 4, 128, 0, stream>>>(cb, norms);
    rvq_main<<<NBLK, NTHR, 0, stream>>>(x, cb, norms, outQ, outIdx, partial);
    rvq_loss<<<1, 256, 0, stream>>>(partial, outLoss);
}